// AbstractNetGen_74801150427859
// MI455X (gfx1250) — compile-verified
//
#include <hip/hip_runtime.h>
#include <math.h>

#define Bb 32
#define Nn 4096
#define DKk 128
#define Hh 128
#define Aa 128
#define Tt 64
#define NCH 16            // context N-chunks
#define CHN (Nn / NCH)    // 256 source rows per chunk
#define NEG_INF_F (-1e9f)

typedef float v2f __attribute__((ext_vector_type(2)));
typedef float v8f __attribute__((ext_vector_type(8)));

// CDNA5 fp32 WMMA: D(16x16) = A(16x4) * B(4x16) + C, exact fp32 math.
__device__ __forceinline__ v8f wmma4(v2f a, v2f b, v8f c) {
    return __builtin_amdgcn_wmma_f32_16x16x4_f32(false, a, false, b, (short)0, c, false, false);
}

__device__ __forceinline__ float sigmf(float x) { return 1.0f / (1.0f + expf(-x)); }

// ---------------------------------------------------------------------------
// K0: init state
// ---------------------------------------------------------------------------
__global__ void init_kernel(const float* __restrict__ h0, const float* __restrict__ c0,
                            float* __restrict__ h, float* __restrict__ c,
                            float* __restrict__ coverage, float* __restrict__ closs_acc) {
    int i = blockIdx.x * 256 + threadIdx.x;
    if (i < Bb * Nn) coverage[i] = 0.0f;
    if (i < Bb * Hh) { h[i] = h0[i]; c[i] = c0[i]; }
    if (i < Bb)      closs_acc[i] = 0.0f;
}

// ---------------------------------------------------------------------------
// K1: z_fea[b,n,:] = z[b,n,:] @ W_z   (M = B*N = 131072, K = 128, N = 128)
// One workgroup = one 16-row M strip; 8 waves cover the 8 N-tiles.
// ---------------------------------------------------------------------------
__global__ void zfea_kernel(const float* __restrict__ z, const float* __restrict__ Wz,
                            float* __restrict__ zfea) {
    __shared__ float zs[16 * 128];  // 8 KB strip of A
    const int m0  = blockIdx.x * 16;
    const int tid = threadIdx.x;
#pragma unroll
    for (int i = 0; i < 8; ++i) {
        int idx = tid + i * 256;
        zs[idx] = z[(size_t)m0 * DKk + idx];
    }
    __syncthreads();

    const int wave = tid >> 5, lane = tid & 31;
    const int l16 = lane & 15;
    const int kb  = (lane >> 4) * 2;     // K sub-offset per A/B fragment layout
    const int n0  = wave * 16;
    v8f acc = {0.f,0.f,0.f,0.f,0.f,0.f,0.f,0.f};
#pragma unroll 4
    for (int ks = 0; ks < 32; ++ks) {
        const int kk = ks * 4 + kb;
        v2f a, bf;
        a.x  = zs[l16 * 128 + kk];
        a.y  = zs[l16 * 128 + kk + 1];
        bf.x = Wz[kk * Aa + n0 + l16];
        bf.y = Wz[(kk + 1) * Aa + n0 + l16];
        acc = wmma4(a, bf, acc);
    }
    const int rbase = (lane >> 4) * 8;
#pragma unroll
    for (int r = 0; r < 8; ++r) {
        int row = m0 + rbase + r;
        zfea[(size_t)row * Aa + n0 + l16] = acc[r];
    }
}

// ---------------------------------------------------------------------------
// K2: one-workgroup LSTM step + dec_fea (all WMMA, latency-critical serial op)
//   gates = [s|h](32x256) @ [W_ih^T;W_hh^T](256x512)
//   h,c   = lstm elementwise
//   dec   = [h|c](32x256) @ W_x(256x128)
// ---------------------------------------------------------------------------
__device__ __forceinline__ float acat(const float* s, int sstride, int use_s,
                                      const float* h, int row, int k) {
    return (k < DKk) ? (use_s ? s[row * sstride + k] : 0.0f)
                     : h[row * Hh + (k - DKk)];
}
__device__ __forceinline__ float bcat(const float* Wih, const float* Whh, int k, int j) {
    return (k < DKk) ? Wih[j * DKk + k] : Whh[j * Hh + (k - DKk)];
}
__device__ __forceinline__ float xcat(const float* h, const float* c, int row, int k) {
    return (k < Hh) ? h[row * Hh + k] : c[row * Hh + (k - Hh)];
}

__global__ void lstm_step_kernel(const float* __restrict__ s, int sstride, int use_s,
                                 float* __restrict__ h, float* __restrict__ c,
                                 const float* __restrict__ Wih, const float* __restrict__ Whh,
                                 const float* __restrict__ bih, const float* __restrict__ bhh,
                                 const float* __restrict__ Wx, float* __restrict__ dec_fea) {
    __shared__ float gl[32 * 512];  // 64 KB gate staging
    const int tid  = threadIdx.x;
    const int wave = tid >> 5, lane = tid & 31;
    const int l16  = lane & 15;
    const int kb   = (lane >> 4) * 2;
    const int rbase = (lane >> 4) * 8;

    // ---- phase A: gates GEMM. wave w owns N-tiles 4w..4w+3, both M-tiles.
    v8f acc[4][2];
#pragma unroll
    for (int p = 0; p < 4; ++p)
#pragma unroll
        for (int m = 0; m < 2; ++m) acc[p][m] = v8f{0.f,0.f,0.f,0.f,0.f,0.f,0.f,0.f};

    for (int ks = 0; ks < 64; ++ks) {
        const int kk = ks * 4 + kb;
        v2f afr[2];
#pragma unroll
        for (int m = 0; m < 2; ++m) {
            int row = m * 16 + l16;
            afr[m].x = acat(s, sstride, use_s, h, row, kk);
            afr[m].y = acat(s, sstride, use_s, h, row, kk + 1);
        }
#pragma unroll
        for (int p = 0; p < 4; ++p) {
            int col = (wave * 4 + p) * 16 + l16;
            v2f bf;
            bf.x = bcat(Wih, Whh, kk, col);
            bf.y = bcat(Wih, Whh, kk + 1, col);
            acc[p][0] = wmma4(afr[0], bf, acc[p][0]);
            acc[p][1] = wmma4(afr[1], bf, acc[p][1]);
        }
    }
#pragma unroll
    for (int p = 0; p < 4; ++p) {
        int n0p = (wave * 4 + p) * 16;
#pragma unroll
        for (int m = 0; m < 2; ++m)
#pragma unroll
            for (int r = 0; r < 8; ++r)
                gl[(m * 16 + rbase + r) * 512 + n0p + l16] = acc[p][m][r];
    }
    __syncthreads();

    // ---- phase B: elementwise LSTM cell (torch gate order i,f,g,o)
#pragma unroll
    for (int i = 0; i < 16; ++i) {
        int idx = tid + i * 256;           // 32*128 = 4096 outputs
        int br = idx >> 7, j = idx & 127;
        float ig = gl[br * 512 + j]        + bih[j]        + bhh[j];
        float fg = gl[br * 512 + 128 + j]  + bih[128 + j]  + bhh[128 + j];
        float gg = gl[br * 512 + 256 + j]  + bih[256 + j]  + bhh[256 + j];
        float og = gl[br * 512 + 384 + j]  + bih[384 + j]  + bhh[384 + j];
        float cold = c[br * Hh + j];
        float cn = sigmf(fg) * cold + sigmf(ig) * tanhf(gg);
        float hn = sigmf(og) * tanhf(cn);
        c[br * Hh + j] = cn;
        h[br * Hh + j] = hn;
    }
    __syncthreads();

    // ---- phase C: dec_fea = [h|c] @ W_x. wave w owns N-tile w, both M-tiles.
    v8f d2[2];
    d2[0] = v8f{0.f,0.f,0.f,0.f,0.f,0.f,0.f,0.f};
    d2[1] = v8f{0.f,0.f,0.f,0.f,0.f,0.f,0.f,0.f};
    const int n0 = wave * 16;
    for (int ks = 0; ks < 64; ++ks) {
        const int kk = ks * 4 + kb;
        v2f afr[2];
#pragma unroll
        for (int m = 0; m < 2; ++m) {
            int row = m * 16 + l16;
            afr[m].x = xcat(h, c, row, kk);
            afr[m].y = xcat(h, c, row, kk + 1);
        }
        v2f bf;
        bf.x = Wx[kk * Aa + n0 + l16];
        bf.y = Wx[(kk + 1) * Aa + n0 + l16];
        d2[0] = wmma4(afr[0], bf, d2[0]);
        d2[1] = wmma4(afr[1], bf, d2[1]);
    }
#pragma unroll
    for (int m = 0; m < 2; ++m)
#pragma unroll
        for (int r = 0; r < 8; ++r)
            dec_fea[(m * 16 + rbase + r) * Aa + n0 + l16] = d2[m][r];
}

// ---------------------------------------------------------------------------
// K3: attention scores. One wave per (b,n): e = v . tanh(z_fea + dec + cov*w_c + b)
// Streams 64 MB of z_fea per step (L2-resident after step 0).
// ---------------------------------------------------------------------------
__global__ void score_kernel(const float* __restrict__ zfea, const float* __restrict__ dec_fea,
                             const float* __restrict__ coverage, const float* __restrict__ wc,
                             const float* __restrict__ battn, const float* __restrict__ vvec,
                             const float* __restrict__ mask, float* __restrict__ e) {
    const int gw   = blockIdx.x * 8 + (threadIdx.x >> 5);
    const int lane = threadIdx.x & 31;
    const int b = gw >> 12;            // N = 4096 waves per batch row
    const int n = gw & (Nn - 1);
    const size_t base = (size_t)b * Nn + n;

    float4 zf  = ((const float4*)zfea)[base * 32 + lane];
    float4 df  = ((const float4*)dec_fea)[b * 32 + lane];
    float4 wc4 = ((const float4*)wc)[lane];
    float4 ba4 = ((const float4*)battn)[lane];
    float4 v4  = ((const float4*)vvec)[lane];
    float  cov = coverage[base];

    float t0 = tanhf(zf.x + df.x + cov * wc4.x + ba4.x);
    float t1 = tanhf(zf.y + df.y + cov * wc4.y + ba4.y);
    float t2 = tanhf(zf.z + df.z + cov * wc4.z + ba4.z);
    float t3 = tanhf(zf.w + df.w + cov * wc4.w + ba4.w);
    float part = t0 * v4.x + t1 * v4.y + t2 * v4.z + t3 * v4.w;

#pragma unroll
    for (int off = 16; off > 0; off >>= 1) part += __shfl_down(part, off);
    if (lane == 0) e[base] = (mask[base] > 0.0f) ? part : NEG_INF_F;
}

// ---------------------------------------------------------------------------
// K4: per-b softmax over N, coverage update + coverage-loss partial (no atomics)
// ---------------------------------------------------------------------------
__global__ void softmax_kernel(const float* __restrict__ e, float* __restrict__ coverage,
                               float* __restrict__ attn_out, float* __restrict__ closs_acc,
                               int t) {
    __shared__ float es[Nn];
    __shared__ float red[256];
    const int b = blockIdx.x, tid = threadIdx.x;
    const float* eb = e + (size_t)b * Nn;

    float lmax = -3.4e38f;
#pragma unroll
    for (int i = 0; i < 16; ++i) {
        int idx = tid + i * 256;
        float vv = eb[idx];
        es[idx] = vv;
        lmax = fmaxf(lmax, vv);
    }
    red[tid] = lmax; __syncthreads();
    for (int sft = 128; sft > 0; sft >>= 1) {
        if (tid < sft) red[tid] = fmaxf(red[tid], red[tid + sft]);
        __syncthreads();
    }
    const float maxv = red[0];
    __syncthreads();

    float lsum = 0.0f;
#pragma unroll
    for (int i = 0; i < 16; ++i) {
        int idx = tid + i * 256;
        float ex = expf(es[idx] - maxv);
        es[idx] = ex;
        lsum += ex;
    }
    red[tid] = lsum; __syncthreads();
    for (int sft = 128; sft > 0; sft >>= 1) {
        if (tid < sft) red[tid] += red[tid + sft];
        __syncthreads();
    }
    const float inv = 1.0f / red[0];
    __syncthreads();

    float part = 0.0f;
    float* covb = coverage + (size_t)b * Nn;
    float* ao   = attn_out + ((size_t)b * Tt + t) * Nn;
#pragma unroll
    for (int i = 0; i < 16; ++i) {
        int idx = tid + i * 256;
        float a_ = es[idx] * inv;
        float co = covb[idx];
        part += fminf(a_, co);          // uses coverage BEFORE update (See et al.)
        covb[idx] = co + a_;
        ao[idx] = a_;
    }
    red[tid] = part; __syncthreads();
    for (int sft = 128; sft > 0; sft >>= 1) {
        if (tid < sft) red[tid] += red[tid + sft];
        __syncthreads();
    }
    if (tid == 0) closs_acc[b] += red[0] * (1.0f / Bb);
}

// ---------------------------------------------------------------------------
// K5a: context partials. Grid = B x NCH workgroups; each WG handles 256 source
// rows with float4 (b128) fully-coalesced loads of z. Deterministic, no atomics.
//   ct_part[b][ch][d] = sum_{n in chunk} attn[b,n] * z[b,n,d]
// ---------------------------------------------------------------------------
__global__ void context_part_kernel(const float* __restrict__ z,
                                    const float* __restrict__ attn_out,
                                    float* __restrict__ ct_part, int t) {
    __shared__ float  attn_s[CHN];
    __shared__ float4 red4[256];
    const int b  = blockIdx.x >> 4;       // NCH = 16
    const int ch = blockIdx.x & (NCH - 1);
    const int tid = threadIdx.x;

    const float* ao = attn_out + ((size_t)b * Tt + t) * Nn + ch * CHN;
    attn_s[tid] = ao[tid];                // CHN == 256 == blockDim
    __syncthreads();

    const int dq = tid & 31;              // float4 quad: dims 4*dq .. 4*dq+3
    const int g  = tid >> 5;              // n-subgroup 0..7 (32 rows each)
    const float4* zb4 = (const float4*)(z + (size_t)b * Nn * DKk) + (size_t)(ch * CHN) * 32;

    float4 acc = {0.f, 0.f, 0.f, 0.f};
    const int nn0 = g * 32;
#pragma unroll 4
    for (int n = nn0; n < nn0 + 32; ++n) {
        float a_ = attn_s[n];
        float4 zv = zb4[(size_t)n * 32 + dq];
        acc.x += a_ * zv.x; acc.y += a_ * zv.y;
        acc.z += a_ * zv.z; acc.w += a_ * zv.w;
    }
    red4[tid] = acc;
    __syncthreads();
    // fixed-order tree reduction over the 8 n-subgroups
    for (int sft = 4; sft >= 1; sft >>= 1) {
        if (g < sft) {
            float4 o = red4[tid + sft * 32];
            acc.x += o.x; acc.y += o.y; acc.z += o.z; acc.w += o.w;
            red4[tid] = acc;
        }
        __syncthreads();
    }
    if (g == 0)
        ((float4*)(ct_part + ((size_t)b * NCH + ch) * DKk))[dq] = acc;
}

// K5b: reduce NCH chunk partials -> text[:,t,:] (which is also s_{t+1})
__global__ void context_final_kernel(const float* __restrict__ ct_part,
                                     float* __restrict__ text_out, int t) {
    int i = blockIdx.x * 256 + threadIdx.x;   // B*DK = 4096
    int b = i >> 7, d = i & 127;
    float ssum = 0.0f;
#pragma unroll
    for (int ch = 0; ch < NCH; ++ch)
        ssum += ct_part[((size_t)b * NCH + ch) * DKk + d];
    text_out[((size_t)b * Tt + t) * DKk + d] = ssum;
}

// ---------------------------------------------------------------------------
// K6: final deterministic coverage-loss sum
// ---------------------------------------------------------------------------
__global__ void closs_final_kernel(const float* __restrict__ closs_acc, float* __restrict__ out) {
    if (threadIdx.x == 0 && blockIdx.x == 0) {
        float ssum = 0.0f;
        for (int i = 0; i < Bb; ++i) ssum += closs_acc[i];
        *out = ssum;
    }
}

// ---------------------------------------------------------------------------
extern "C" void kernel_launch(void* const* d_in, const int* in_sizes, int n_in,
                              void* d_out, int out_size, void* d_ws, size_t ws_size,
                              hipStream_t stream) {
    (void)in_sizes; (void)n_in; (void)out_size; (void)ws_size;
    const float* z     = (const float*)d_in[0];
    const float* mask  = (const float*)d_in[1];
    const float* h0    = (const float*)d_in[2];
    const float* c0    = (const float*)d_in[3];
    const float* Wih   = (const float*)d_in[4];
    const float* Whh   = (const float*)d_in[5];
    const float* bih   = (const float*)d_in[6];
    const float* bhh   = (const float*)d_in[7];
    const float* Wx    = (const float*)d_in[8];
    const float* Wz    = (const float*)d_in[9];
    const float* wc    = (const float*)d_in[10];
    const float* battn = (const float*)d_in[11];
    const float* vvec  = (const float*)d_in[12];
    // d_in[13] = n_node (device scalar); loop trip count must be host-known
    // and is 64 in the reference configuration.

    float* text  = (float*)d_out;                       // [B,T,DK]
    float* attn  = text + (size_t)Bb * Tt * DKk;        // [B,T,N]
    float* closs = attn + (size_t)Bb * Tt * Nn;         // scalar

    float* ws       = (float*)d_ws;
    float* zfea     = ws;  ws += (size_t)Bb * Nn * Aa;  // 64 MB (L2-resident)
    float* coverage = ws;  ws += (size_t)Bb * Nn;
    float* h        = ws;  ws += Bb * Hh;
    float* c        = ws;  ws += Bb * Hh;
    float* dec      = ws;  ws += Bb * Aa;
    float* e        = ws;  ws += (size_t)Bb * Nn;
    float* ctp      = ws;  ws += (size_t)Bb * NCH * DKk;
    float* clacc    = ws;  ws += Bb;

    init_kernel<<<512, 256, 0, stream>>>(h0, c0, h, c, coverage, clacc);
    zfea_kernel<<<(Bb * Nn) / 16, 256, 0, stream>>>(z, Wz, zfea);

    for (int t = 0; t < Tt; ++t) {
        const float* s = (t == 0) ? text : (text + (size_t)(t - 1) * DKk);
        const int sstride = Tt * DKk;   // text is [B,T,DK]: batch stride T*DK
        lstm_step_kernel<<<1, 256, 0, stream>>>(s, sstride, (t != 0) ? 1 : 0,
                                                h, c, Wih, Whh, bih, bhh, Wx, dec);
        score_kernel<<<(Bb * Nn) / 8, 256, 0, stream>>>(zfea, dec, coverage, wc,
                                                        battn, vvec, mask, e);
        softmax_kernel<<<Bb, 256, 0, stream>>>(e, coverage, attn, clacc, t);
        context_part_kernel<<<Bb * NCH, 256, 0, stream>>>(z, attn, ctp, t);
        context_final_kernel<<<(Bb * DKk) / 256, 256, 0, stream>>>(ctp, text, t);
    }
    closs_final_kernel<<<1, 32, 0, stream>>>(clacc, closs);
}